// LSTM_13520557048426
// MI455X (gfx1250) — compile-verified
//
#include <hip/hip_runtime.h>

namespace {

constexpr int BATCH = 4096;
constexpr int T     = 1024;
constexpr int TC    = 32;                 // time-chunk staged in LDS
constexpr int NC    = T / TC;             // 32 chunks
constexpr int XROW  = 2 * TC + 2;         // floats per staged row (+8B pad: conflict-free b64 reads)
constexpr int WAVES_PER_BLOCK = 4;
constexpr int ROWS_PER_WAVE   = 16;
constexpr int LDS_PER_WAVE = 2 * 16 * XROW;   // double-buffered x stage (floats)

typedef __attribute__((ext_vector_type(2))) float v2f;
typedef __attribute__((ext_vector_type(4))) float v4f;
typedef __attribute__((ext_vector_type(8))) float v8f;

__device__ __forceinline__ float fast_tanh(float x) {
#if __has_builtin(__builtin_amdgcn_tanhf)
  return __builtin_amdgcn_tanhf(x);          // v_tanh_f32 (CDNA5 TRANS op)
#else
  float ax = __builtin_fabsf(x);
  float e  = __builtin_amdgcn_exp2f(-2.8853900817779268f * ax);
  float r  = 1.0f - 2.0f * e * __builtin_amdgcn_rcpf(1.0f + e);
  return __builtin_copysignf(r, x);
#endif
}

// 16B-per-lane async copy global -> LDS (CDNA5 async path), sync fallback.
// Signature (probe-verified): (v4i AS1*, v4i AS3*, imm offset, imm cpol)
__device__ __forceinline__ void async_cp16(const float* g, float* l) {
#if __has_builtin(__builtin_amdgcn_global_load_async_to_lds_b128)
  typedef int v4i_ __attribute__((vector_size(16)));
  typedef __attribute__((address_space(1))) v4i_* gp_t;
  typedef __attribute__((address_space(3))) v4i_* lp_t;
  __builtin_amdgcn_global_load_async_to_lds_b128((gp_t)g, (lp_t)l, 0, 0);
#else
  *(v4f*)l = *(const v4f*)g;
#endif
}

} // namespace

__global__ __launch_bounds__(WAVES_PER_BLOCK * 32)
void lstm_wmma_kernel(const float* __restrict__ x,      // [B,T,2]
                      const float* __restrict__ W_ih,   // [8,2]
                      const float* __restrict__ W_hh,   // [8,2]
                      const float* __restrict__ b_ih,   // [8]
                      const float* __restrict__ b_hh,   // [8]
                      const float* __restrict__ W_out,  // [1,2]
                      const float* __restrict__ b_out,  // [1]
                      const int*   __restrict__ slice_start,
                      float*       __restrict__ out) {  // [B, T-ss, 1]
  __shared__ float lds[WAVES_PER_BLOCK * LDS_PER_WAVE];

  const int tid   = threadIdx.x;
  const int lane  = tid & 31;
  const int wv    = tid >> 5;
  const int lo    = lane & 15;
  const int hi    = lane >> 4;              // 0: lanes 0-15, 1: lanes 16-31
  const bool loHalf = (lane < 16);
  const int row0  = (blockIdx.x * WAVES_PER_BLOCK + wv) * ROWS_PER_WAVE;

  float* wbase = lds + wv * LDS_PER_WAVE;
  float* xbuf0 = wbase;
  float* xbuf1 = wbase + 16 * XROW;

  // ---- transposed-product formulation ----------------------------------
  // gates^T(16x16) = A x B + C with
  //   A (16x4, constant) : row n = gate, A(n,0..1)=W_hh[n,:], A(n,2..3)=W_ih[n,:]
  //   B (4x16, dynamic)  : col m = batch row, B(:,m) = [h0,h1,x0,x1]
  //   C                  : bias broadcast per gate row
  // D VGPR r = gate r across batch lanes 0-15 (rows 8..15 / lanes 16-31: pad).
  // Sigmoid gates folded to tanh: rows {0,1,2,3,6,7} of A and C scaled by 0.5,
  // then sigmoid(z) = 0.5*tanh(z/2)+0.5 applied per-VGPR (compile-time).
  const int  n   = lo;
  const bool isG = (n == 4) || (n == 5);
  const float gsc = (n < 8) ? (isG ? 1.0f : 0.5f) : 0.0f;
  float a0 = 0.0f, a1 = 0.0f;
  if (n < 8) {
    const float* W = loHalf ? W_hh : W_ih;  // lanes 0-15: K0,K1 ; 16-31: K2,K3
    a0 = W[2 * n + 0] * gsc;
    a1 = W[2 * n + 1] * gsc;
  }
  v2f amat = {a0, a1};

  v8f cbias;
#pragma unroll
  for (int r = 0; r < 8; ++r) {
    const float s = (r == 4 || r == 5) ? 1.0f : 0.5f;
    cbias[r] = loHalf ? (b_ih[r] + b_hh[r]) * s : 0.0f;   // rows 8..15 pad = 0
  }

  const float wo0 = W_out[0], wo1 = W_out[1], bo = b_out[0];
  const int ss = slice_start[0];
  const int OT = T - ss;
  // hoisted output row pointer, pre-biased by -ss so index is just t
  float* outp = out + (size_t)(row0 + lane) * OT - ss;

  // ---- coalesced async staging of x chunks -----------------------------
  const float* grow = x + (size_t)(row0 + hi) * (2 * T) + lo * 4;
  auto issue = [&](float* buf, int t0) {
    const float* g = grow + 2 * t0;
    float* l = buf + hi * XROW + lo * 4;
#pragma unroll
    for (int k = 0; k < 8; ++k)
      async_cp16(g + k * (4 * T), l + k * (2 * XROW));
  };

  float h0 = 0.0f, h1 = 0.0f, c0 = 0.0f, c1 = 0.0f;

  // one LSTM cell step for all 16 rows of this wave
  auto step = [&](v2f xv) {
    v2f bdyn = {loHalf ? h0 : xv[0], loHalf ? h1 : xv[1]};
    v8f d = __builtin_amdgcn_wmma_f32_16x16x4_f32(
        false, amat, false, bdyn, (short)0, cbias, false, false);
    float ga[8];
#pragma unroll
    for (int r = 0; r < 8; ++r) {
      const float th = fast_tanh(d[r]);
      ga[r] = (r == 4 || r == 5) ? th : __builtin_fmaf(th, 0.5f, 0.5f);
    }
    c0 = __builtin_fmaf(ga[2], c0, ga[0] * ga[4]);
    c1 = __builtin_fmaf(ga[3], c1, ga[1] * ga[5]);
    h0 = ga[6] * fast_tanh(c0);
    h1 = ga[7] * fast_tanh(c1);
  };

  issue(xbuf0, 0);

  for (int ch = 0; ch < NC; ++ch) {
    if (ch + 1 < NC) {
      issue((ch & 1) ? xbuf0 : xbuf1, (ch + 1) * TC);   // prefetch next chunk
      asm volatile("s_wait_asynccnt 8" ::: "memory");   // oldest 8 (this chunk) done
    } else {
      asm volatile("s_wait_asynccnt 0" ::: "memory");
    }
    const float* xwr = ((ch & 1) ? xbuf1 : xbuf0) + lo * XROW;
    const int tbase = ch * TC;

    v2f xv = *(const v2f*)xwr;                 // x_t for first step of chunk
    if (tbase + TC <= ss) {
      // ---- store-free phase (uniform): pure recurrence ----
#pragma unroll 8
      for (int tt = 0; tt < TC; ++tt) {
        v2f xn = xv;
        if (tt + 1 < TC) xn = *(const v2f*)(xwr + 2 * (tt + 1));  // pipelined
        step(xv);
        xv = xn;
      }
    } else {
      // ---- output phase (last chunks / boundary chunk) ----
#pragma unroll 8
      for (int tt = 0; tt < TC; ++tt) {
        v2f xn = xv;
        if (tt + 1 < TC) xn = *(const v2f*)(xwr + 2 * (tt + 1));  // pipelined
        step(xv);
        const int t = tbase + tt;
        if (t >= ss && loHalf)
          outp[t] = __builtin_fmaf(h0, wo0, __builtin_fmaf(h1, wo1, bo));
        xv = xn;
      }
    }
  }
}

extern "C" void kernel_launch(void* const* d_in, const int* in_sizes, int n_in,
                              void* d_out, int out_size, void* d_ws, size_t ws_size,
                              hipStream_t stream) {
  (void)in_sizes; (void)n_in; (void)out_size; (void)d_ws; (void)ws_size;
  const float* x     = (const float*)d_in[0];
  const float* W_ih  = (const float*)d_in[1];
  const float* W_hh  = (const float*)d_in[2];
  const float* b_ih  = (const float*)d_in[3];
  const float* b_hh  = (const float*)d_in[4];
  const float* W_out = (const float*)d_in[5];
  const float* b_out = (const float*)d_in[6];
  const int*   ss    = (const int*)d_in[7];

  dim3 grid(BATCH / (WAVES_PER_BLOCK * ROWS_PER_WAVE));   // 64 blocks
  dim3 block(WAVES_PER_BLOCK * 32);                       // 128 threads (4 waves)
  lstm_wmma_kernel<<<grid, block, 0, stream>>>(
      x, W_ih, W_hh, b_ih, b_hh, W_out, b_out, ss, (float*)d_out);
}